// PopSpikeEncoderDeterministic_18047452578017
// MI455X (gfx1250) — compile-verified
//
#include <hip/hip_runtime.h>
#include <hip/hip_bf16.h>

// PopSpikeEncoderDeterministic for MI455X (gfx1250) — final form.
//
// Cost model:
//   obs  [1024, 256] f32           (~1 MiB, L2-resident across all blocks)
//   mean [1, 256, 32] f32 = [8192] (32 KiB, L2-resident)
//   std  [1, 256, 32] f32 = [8192]
//   out  [16, 1024, 8192] f32      (512 MiB -> store-bandwidth bound,
//                                   ~22.5 us floor @ 23.3 TB/s)
//
// Purely store-bound and contraction-free: WMMA is mathematically
// inapplicable. The CDNA5 feature that matters here is cache control:
// nontemporal B128 streaming stores (output stream is 2.7x the 192 MB L2,
// so NT keeps the L2-resident obs/mean/std working set alive for all blocks).
//
// Each thread owns 8 contiguous encoder elements (one shared obs scalar;
// an aligned 8-run never crosses a POP_DIM=32 boundary): computes the
// Gaussian activation once in registers (v_rcp + ocml expf), runs the
// 16-step soft-reset IF recurrence fully unrolled in registers, and emits
// 2 NT B128 stores per timestep (32 in flight per thread total).

typedef float v4f __attribute__((ext_vector_type(4)));

#define OBS_DIM   256
#define POP_DIM   32
#define ENC       8192        // OBS_DIM * POP_DIM
#define SPIKE_TS  16
#define BATCH     1024
#define V_TH      0.999f

__global__ __launch_bounds__(256) void popspike_if_kernel(
    const float* __restrict__ obs,    // [BATCH, OBS_DIM]
    const float* __restrict__ mean,   // [ENC]
    const float* __restrict__ stdv,   // [ENC]
    float* __restrict__ out)          // [SPIKE_TS, BATCH, ENC]
{
    const int tid = blockIdx.x * blockDim.x + threadIdx.x;   // 0 .. BATCH*ENC/8-1
    const int e8  = tid << 3;              // flat element index (multiple of 8)
    const int b   = e8 >> 13;              // / ENC
    const int e   = e8 & (ENC - 1);        // encoder index, multiple of 8
    const int o   = e >> 5;                // / POP_DIM (same o for all 8 lanes)

    // One scalar obs value shared by this thread's 8 encoder lanes.
    const float x = obs[(b << 8) + o];

    // 16B-aligned vector loads of the per-element Gaussian params.
    const v4f m0 = *(const v4f*)(mean + e);
    const v4f m1 = *(const v4f*)(mean + e + 4);
    const v4f s0 = *(const v4f*)(stdv + e);
    const v4f s1 = *(const v4f*)(stdv + e + 4);

    // pop_act = exp(-0.5 * (x - m)^2 / s^2), computed once, lives in regs.
    // v_rcp_f32 (<=1 ulp) instead of IEEE divide: perturbs the exponent arg
    // by <= |arg|*1.2e-7; any element that can spike has |arg| <= ~2.8, so
    // |da/a| <= ~3e-7 -- below ocml-expf-vs-XLA-exp noise.
    float a[8], v[8];
#pragma unroll
    for (int i = 0; i < 4; ++i) {
        const float d0   = x - m0[i];
        const float d1   = x - m1[i];
        const float inv0 = -0.5f * __builtin_amdgcn_rcpf(s0[i] * s0[i]);
        const float inv1 = -0.5f * __builtin_amdgcn_rcpf(s1[i] * s1[i]);
        a[i]     = expf(d0 * d0 * inv0);
        a[i + 4] = expf(d1 * d1 * inv1);
        v[i]     = 0.0f;
        v[i + 4] = 0.0f;
    }

    // Soft-reset IF neuron, T=16, same input each step. v stays in registers.
    float* base = out + (b << 13) + e;     // plane t=0
#pragma unroll
    for (int t = 0; t < SPIKE_TS; ++t) {
        v4f sp0, sp1;
#pragma unroll
        for (int i = 0; i < 4; ++i) {
            v[i] += a[i];
            sp0[i] = (v[i] >= V_TH) ? 1.0f : 0.0f;
            v[i] -= sp0[i] * V_TH;         // soft reset (detached in fwd)

            v[i + 4] += a[i + 4];
            sp1[i] = (v[i + 4] >= V_TH) ? 1.0f : 0.0f;
            v[i + 4] -= sp1[i] * V_TH;
        }
        // Streaming NT stores: 512 MiB output must not thrash the 192 MB L2.
        __builtin_nontemporal_store(sp0, (v4f*)base);
        __builtin_nontemporal_store(sp1, (v4f*)(base + 4));
        base += BATCH * ENC;               // next timestep plane (+32 MiB)
    }
}

extern "C" void kernel_launch(void* const* d_in, const int* in_sizes, int n_in,
                              void* d_out, int out_size, void* d_ws, size_t ws_size,
                              hipStream_t stream) {
    const float* obs  = (const float*)d_in[0];   // [1024*256]
    const float* mean = (const float*)d_in[1];   // [8192]
    const float* stdv = (const float*)d_in[2];   // [8192]
    float* out = (float*)d_out;                  // [16*1024*8192]

    const int total_threads = BATCH * ENC / 8;   // 1,048,576
    const int block = 256;                       // 8 waves per WG on wave32
    const int grid  = total_threads / block;     // 4096 blocks

    popspike_if_kernel<<<grid, block, 0, stream>>>(obs, mean, stdv, out);
}